// AttentionModel_23536420782781
// MI455X (gfx1250) — compile-verified
//
#include <hip/hip_runtime.h>
#include <hip/hip_bf16.h>

typedef __attribute__((ext_vector_type(2))) float v2f;
typedef __attribute__((ext_vector_type(8))) float v8f;

#define LOG2E 1.4426950408889634f

__device__ __forceinline__ float fast_tanh(float x) {
  // tanh(x) = 1 - 2/(e^{2x}+1); v_exp_f32 + v_rcp_f32, no div-fixup sequence.
  // x>>0: e->inf, rcp->0 -> +1 ; x<<0: e->0, rcp(1)=1 -> -1. Saturation OK.
  float e = __builtin_amdgcn_exp2f(x * (2.0f * LOG2E));
  return 1.0f - 2.0f * __builtin_amdgcn_rcpf(e + 1.0f);
}

__device__ __forceinline__ float fast_exp(float x) {
  return __builtin_amdgcn_exp2f(x * LOG2E);
}

// ---------------------------------------------------------------------------
// fp32 GEMM via V_WMMA_F32_16X16X4_F32, row-major output.
// Y[b][M,N] = X[b][M,K] @ W[b][K,N] (+ bias[N]); one wave per 16x16 tile.
// ---------------------------------------------------------------------------
__global__ __launch_bounds__(256)
void gemm_wmma_f32(const float* __restrict__ X,
                   const float* __restrict__ Wm,
                   const float* __restrict__ bias,
                   float* __restrict__ Y,
                   int M, int K, int N,
                   long sX, long sW, long sY) {
  const long b = blockIdx.y;
  X  += b * sX;
  Wm += b * sW;
  Y  += b * sY;

  const int wave = blockIdx.x * (blockDim.x >> 5) + (threadIdx.x >> 5);
  const int lane = threadIdx.x & 31;
  const int tilesN = N >> 4;
  const int tM = wave / tilesN;
  const int tN = wave - tM * tilesN;
  if ((tM << 4) >= M) return;   // whole wave exits together -> EXEC stays full

  const int  row0 = tM << 4;
  const int  col0 = tN << 4;
  const bool hi   = lane >= 16;
  const int  l15  = lane & 15;
  const int  mrow = row0 + l15;
  const int  col  = col0 + l15;

  v8f acc = {};
  const float* xp = X + (long)mrow * K + (hi ? 2 : 0);
  const float* wp = Wm + (long)(hi ? 2 : 0) * N + col;

#pragma unroll 4
  for (int k0 = 0; k0 < K; k0 += 4) {
    v2f a, bm;
    a.x  = xp[k0];
    a.y  = xp[k0 + 1];
    bm.x = wp[(long)k0 * N];
    bm.y = wp[(long)(k0 + 1) * N];
    acc = __builtin_amdgcn_wmma_f32_16x16x4_f32(
        false, a, false, bm, (short)0, acc, false, false);
  }

  const float bb = bias ? bias[col] : 0.0f;
  const int rbase = row0 + (hi ? 8 : 0);
#pragma unroll
  for (int r = 0; r < 8; ++r)
    Y[(long)(rbase + r) * N + col] = acc[r] + bb;
}

// ---------------------------------------------------------------------------
// Same GEMM but writes the TRANSPOSED output YT (N x M). Each lane's 8 acc
// values are 8 consecutive M-rows at one N-column -> in transposed layout
// they are 8 consecutive floats: two global_store_b128 per lane.
// ---------------------------------------------------------------------------
__global__ __launch_bounds__(256)
void gemm_wmma_f32_tout(const float* __restrict__ X,
                        const float* __restrict__ Wm,
                        const float* __restrict__ bias,
                        float* __restrict__ YT,   // (N, M)
                        int M, int K, int N) {
  const int wave = blockIdx.x * (blockDim.x >> 5) + (threadIdx.x >> 5);
  const int lane = threadIdx.x & 31;
  const int tilesN = N >> 4;
  const int tM = wave / tilesN;
  const int tN = wave - tM * tilesN;
  if ((tM << 4) >= M) return;

  const int  row0 = tM << 4;
  const int  col0 = tN << 4;
  const bool hi   = lane >= 16;
  const int  l15  = lane & 15;
  const int  mrow = row0 + l15;
  const int  col  = col0 + l15;

  v8f acc = {};
  const float* xp = X + (long)mrow * K + (hi ? 2 : 0);
  const float* wp = Wm + (long)(hi ? 2 : 0) * N + col;

#pragma unroll 4
  for (int k0 = 0; k0 < K; k0 += 4) {
    v2f a, bm;
    a.x  = xp[k0];
    a.y  = xp[k0 + 1];
    bm.x = wp[(long)k0 * N];
    bm.y = wp[(long)(k0 + 1) * N];
    acc = __builtin_amdgcn_wmma_f32_16x16x4_f32(
        false, a, false, bm, (short)0, acc, false, false);
  }

  const float bb = bias ? bias[col] : 0.0f;
  const int rbase = row0 + (hi ? 8 : 0);          // multiple of 8 -> 16B aligned
  float* yt = YT + (long)col * M + rbase;
  float4 lo = make_float4(acc[0] + bb, acc[1] + bb, acc[2] + bb, acc[3] + bb);
  float4 hi4 = make_float4(acc[4] + bb, acc[5] + bb, acc[6] + bb, acc[7] + bb);
  *(float4*)(yt)     = lo;
  *(float4*)(yt + 4) = hi4;
}

// ---------------------------------------------------------------------------
// Fused score + softmax, transposed-vc version. One block (256 thr) per (b,s).
// Thread owns 4 context columns: c = tid + 256j. H-reduction is pure register
// accumulation; vcT loads are lane-coalesced; vs/v come from LDS broadcasts.
// This is the transcendental-bound stage: inner loop ~= 4x(v_exp+v_rcp+4 VALU).
// ---------------------------------------------------------------------------
__global__ __launch_bounds__(256)
void scores_softmax_t(const float* __restrict__ vs,     // (B,S,H)
                      const float* __restrict__ vcT,    // (H, B*C)
                      const float* __restrict__ vvec,   // (H)
                      const float* __restrict__ bv_ptr, // scalar
                      float* __restrict__ att,          // (B,S,C)
                      int S, int C, int H, int BC) {
  const int b   = blockIdx.y;
  const int s   = blockIdx.x;
  const int tid = threadIdx.x;
  const int lane = tid & 31;
  const int wv   = tid >> 5;

  __shared__ float s_vs[128];
  __shared__ float s_v[128];
  __shared__ float s_red[8];

  if (tid < H) {
    s_vs[tid] = vs[((long)b * S + s) * H + tid];
    s_v[tid]  = vvec[tid];
  }
  __syncthreads();

  const float* base = vcT + (long)b * C + tid;   // columns c = tid + {0,256,512,768}
  float acc0 = 0.f, acc1 = 0.f, acc2 = 0.f, acc3 = 0.f;
  for (int h = 0; h < 128; ++h) {
    const float* row = base + (long)h * BC;
    const float t  = s_vs[h];       // broadcast LDS read
    const float vv = s_v[h];        // broadcast LDS read
    acc0 += vv * fast_tanh(t + row[0]);
    acc1 += vv * fast_tanh(t + row[256]);
    acc2 += vv * fast_tanh(t + row[512]);
    acc3 += vv * fast_tanh(t + row[768]);
  }
  const float bvv = bv_ptr[0];
  acc0 += bvv; acc1 += bvv; acc2 += bvv; acc3 += bvv;

  // --- block max ---
  float m = fmaxf(fmaxf(acc0, acc1), fmaxf(acc2, acc3));
#pragma unroll
  for (int off = 16; off; off >>= 1) m = fmaxf(m, __shfl_xor(m, off, 32));
  if (lane == 0) s_red[wv] = m;
  __syncthreads();
  m = s_red[0];
#pragma unroll
  for (int w = 1; w < 8; ++w) m = fmaxf(m, s_red[w]);
  __syncthreads();   // before s_red is reused for the sum

  // --- exp + block sum ---
  const float e0 = fast_exp(acc0 - m);
  const float e1 = fast_exp(acc1 - m);
  const float e2 = fast_exp(acc2 - m);
  const float e3 = fast_exp(acc3 - m);
  float sum = (e0 + e1) + (e2 + e3);
#pragma unroll
  for (int off = 16; off; off >>= 1) sum += __shfl_xor(sum, off, 32);
  if (lane == 0) s_red[wv] = sum;
  __syncthreads();
  float tot = 0.f;
#pragma unroll
  for (int w = 0; w < 8; ++w) tot += s_red[w];
  const float inv = __builtin_amdgcn_rcpf(tot);

  float* arow = att + ((long)b * S + s) * C + tid;
  arow[0]   = e0 * inv;
  arow[256] = e1 * inv;
  arow[512] = e2 * inv;
  arow[768] = e3 * inv;
}

// ---------------------------------------------------------------------------
extern "C" void kernel_launch(void* const* d_in, const int* in_sizes, int n_in,
                              void* d_out, int out_size, void* d_ws, size_t ws_size,
                              hipStream_t stream) {
  const int B = 4, S = 256, C = 1024, D = 512, H = 128, K = 512;

  const float* context = (const float*)d_in[0];  // (B,C,D)
  const float* state   = (const float*)d_in[1];  // (B,S,K)
  const float* W       = (const float*)d_in[2];  // (K,H)
  const float* bW      = (const float*)d_in[3];  // (H)
  const float* U       = (const float*)d_in[4];  // (D,H)
  const float* bU      = (const float*)d_in[5];  // (H)
  const float* v       = (const float*)d_in[6];  // (H)
  const float* bv      = (const float*)d_in[7];  // scalar (device)

  float* out_ctx = (float*)d_out;                       // (B,S,D)
  float* att     = (float*)d_out + (size_t)B * S * D;   // (B,S,C)

  float* vs  = (float*)d_ws;                  // (B,S,H)   512 KB
  float* vcT = vs + (size_t)B * S * H;        // (H, B*C)  2 MB, transposed

  // 1) v_state = state @ W + bW   (M=1024, K=512, N=128), row-major out
  {
    const int M = B * S, N = H;
    const int waves = (M / 16) * (N / 16);
    gemm_wmma_f32<<<dim3((waves + 7) / 8, 1), 256, 0, stream>>>(
        state, W, bW, vs, M, K, N, 0, 0, 0);
  }
  // 2) v_context^T = (context @ U + bU)^T   (M=4096, K=512, N=128) -> (128, 4096)
  {
    const int M = B * C, N = H;
    const int waves = (M / 16) * (N / 16);
    gemm_wmma_f32_tout<<<dim3((waves + 7) / 8, 1), 256, 0, stream>>>(
        context, U, bU, vcT, M, K, N);
  }
  // 3) fused scores + softmax -> att
  scores_softmax_t<<<dim3(S, B), 256, 0, stream>>>(
      vs, vcT, v, bv, att, S, C, H, B * C);

  // 4) out_ctx[b] = att[b] @ context[b]   (M=256, K=1024, N=512, batched)
  {
    const int M = S, Kc = C, N = D;
    const int waves = (M / 16) * (N / 16);
    gemm_wmma_f32<<<dim3((waves + 7) / 8, B), 256, 0, stream>>>(
        att, context, nullptr, out_ctx, M, Kc, N,
        (long)S * C, (long)C * D, (long)S * D);
  }
}